// MultiheadSelfAttentionVarSizedElementReduce_41575283426057
// MI455X (gfx1250) — compile-verified
//
#include <hip/hip_runtime.h>

// MI455X / gfx1250, wave32. Fused pipeline for segment-softmax attention pooling.
//
// Stage A: scores[v][h] = sum_d (emb[v,:] @ W_v)[h*64+d]^2      (fp32 WMMA GEMM)
// Stage B: segment softmax over sorted map + weighted pooling of raw emb
// Stage C: out = pooled @ W_out                                  (fp32 WMMA GEMM)
//
// Workspace layout (floats): [ scores: V*H | pooled: S*HID ]  ~18.3 MB total.

typedef __attribute__((ext_vector_type(2))) float v2f;
typedef __attribute__((ext_vector_type(8))) float v8f;

#define V_N   500000
#define D_N   64
#define H_N   4
#define HID_N 256
#define S_N   10000
#define OUT_N 128
#define HEAD_DIM 64  // HID_N / H_N

// ---------------------------------------------------------------------------
// Kernel A: per-wave 16-row tile of V. values = emb @ W_v, reduced immediately
// to per-head sum of squares (full [V,256] values matrix never hits HBM).
//
// WMMA f32 16x16x4 operand layout (wave32):
//   A (16x4):  lane l<16: a = {A[l][4k+0], A[l][4k+1]};  lane>=16: K offset +2
//   B (4x16):  lane l<16: b = {B[4k+0][col], B[4k+1][col]}; lane>=16: K +2
//   C/D:       VGPR r: lanes 0-15 -> row r, lanes 16-31 -> row r+8
// ---------------------------------------------------------------------------
__global__ __launch_bounds__(128) void mha_scores_kernel(
    const float* __restrict__ emb, const float* __restrict__ Wv,
    float* __restrict__ scores) {
  // Per-head transposed weight slice: WT[n*64 + k] = Wv[k*HID + h*64 + n]
  // -> B fragment {WT[col*64+kr], WT[col*64+kr+1]} is one 8B-aligned ds_load_b64.
  __shared__ float WT[HEAD_DIM * D_N];  // 16 KB

  const int tid  = threadIdx.x;
  const int wave = tid >> 5;
  const int lane = tid & 31;
  const int half = lane >> 4;  // 0: lanes 0-15, 1: lanes 16-31
  const int l    = lane & 15;
  const int row0 = blockIdx.x * 64 + wave * 16;
  const bool active = (row0 + 16) <= V_N;  // wave-uniform (V % 16 == 0)

  // A fragments for all 16 K-steps (reused across every head / N-tile).
  v2f a[16];
  if (active) {
    const float* arow = emb + (size_t)(row0 + l) * D_N;
#pragma unroll
    for (int k = 0; k < 16; ++k)
      a[k] = *reinterpret_cast<const v2f*>(arow + 4 * k + 2 * half);
  }

  for (int h = 0; h < H_N; ++h) {
    __syncthreads();  // retire previous head's WT reads
    for (int i = tid; i < HEAD_DIM * D_N; i += 128) {
      const int n = i >> 6, k = i & 63;
      WT[n * D_N + k] = Wv[(size_t)k * HID_N + h * HEAD_DIM + n];
    }
    __syncthreads();
    if (!active) continue;

    float accsq[8];
#pragma unroll
    for (int r = 0; r < 8; ++r) accsq[r] = 0.f;

    for (int nt = 0; nt < 4; ++nt) {
      const int ncol = nt * 16 + l;  // column within this head
      v8f c = {0.f, 0.f, 0.f, 0.f, 0.f, 0.f, 0.f, 0.f};
#pragma unroll
      for (int k = 0; k < 16; ++k) {
        const v2f b =
            *reinterpret_cast<const v2f*>(&WT[ncol * D_N + 4 * k + 2 * half]);
        c = __builtin_amdgcn_wmma_f32_16x16x4_f32(
            /*neg_a=*/false, a[k], /*neg_b=*/false, b,
            /*c_mod=*/(short)0, c, /*reuse_a=*/false, /*reuse_b=*/false);
      }
#pragma unroll
      for (int r = 0; r < 8; ++r) accsq[r] += c[r] * c[r];
    }

    // Butterfly-reduce each accumulator register across its 16-lane half:
    // lanes 0-15 hold rows r, lanes 16-31 hold rows r+8.
#pragma unroll
    for (int r = 0; r < 8; ++r) {
      float s = accsq[r];
      s += __shfl_xor(s, 1);
      s += __shfl_xor(s, 2);
      s += __shfl_xor(s, 4);
      s += __shfl_xor(s, 8);
      if (l == 0)
        scores[(size_t)(row0 + half * 8 + r) * H_N + h] = s;
    }
  }
}

// ---------------------------------------------------------------------------
// Kernel B: one block per sample. Map is sorted -> binary search the segment,
// then max / sum-exp / weighted pooling entirely inside the block.
// Thread t owns (head = t>>6, dim = t&63) of the pooled output.
// ---------------------------------------------------------------------------
__global__ __launch_bounds__(256) void mha_pool_kernel(
    const float* __restrict__ emb, const float* __restrict__ scores,
    const int* __restrict__ seg, float* __restrict__ pooled) {
  const int s   = blockIdx.x;
  const int tid = threadIdx.x;

  // lower_bound(seg, s) and lower_bound(seg, s+1)
  int lo = 0, hi = V_N;
  while (lo < hi) { const int mid = (lo + hi) >> 1; if (seg[mid] < s) lo = mid + 1; else hi = mid; }
  const int start = lo;
  hi = V_N;
  while (lo < hi) { const int mid = (lo + hi) >> 1; if (seg[mid] < s + 1) lo = mid + 1; else hi = mid; }
  const int end = lo;

  __shared__ float red[256 * H_N];
  __shared__ float smax[H_N], sinv[H_N];
  __shared__ float probs[256 * H_N];

  // ---- phase 1: per-head max --------------------------------------------
  float m[H_N];
#pragma unroll
  for (int h = 0; h < H_N; ++h) m[h] = -__builtin_inff();
  for (int v = start + tid; v < end; v += 256) {
#pragma unroll
    for (int h = 0; h < H_N; ++h)
      m[h] = fmaxf(m[h], scores[(size_t)v * H_N + h]);
  }
#pragma unroll
  for (int h = 0; h < H_N; ++h) red[tid * H_N + h] = m[h];
  __syncthreads();
  for (int strd = 128; strd > 0; strd >>= 1) {
    if (tid < strd) {
#pragma unroll
      for (int h = 0; h < H_N; ++h)
        red[tid * H_N + h] = fmaxf(red[tid * H_N + h], red[(tid + strd) * H_N + h]);
    }
    __syncthreads();
  }
  if (tid < H_N) smax[tid] = red[tid];
  __syncthreads();

  // ---- phase 2: per-head sum of exp -------------------------------------
  float e[H_N];
#pragma unroll
  for (int h = 0; h < H_N; ++h) e[h] = 0.f;
  for (int v = start + tid; v < end; v += 256) {
#pragma unroll
    for (int h = 0; h < H_N; ++h)
      e[h] += expf(scores[(size_t)v * H_N + h] - smax[h]);
  }
#pragma unroll
  for (int h = 0; h < H_N; ++h) red[tid * H_N + h] = e[h];
  __syncthreads();
  for (int strd = 128; strd > 0; strd >>= 1) {
    if (tid < strd) {
#pragma unroll
      for (int h = 0; h < H_N; ++h)
        red[tid * H_N + h] += red[(tid + strd) * H_N + h];
    }
    __syncthreads();
  }
  if (tid < H_N) sinv[tid] = 1.f / red[tid];
  __syncthreads();

  // ---- phase 3: prob-weighted pooling of raw embeddings ------------------
  const int h = tid >> 6;
  const int d = tid & 63;
  float acc = 0.f;
  for (int cb = start; cb < end; cb += 256) {
    const int cnt = min(256, end - cb);
    if (tid < cnt) {
#pragma unroll
      for (int hh = 0; hh < H_N; ++hh)
        probs[tid * H_N + hh] =
            expf(scores[(size_t)(cb + tid) * H_N + hh] - smax[hh]) * sinv[hh];
    }
    __syncthreads();
    for (int j = 0; j < cnt; ++j)
      acc += probs[j * H_N + h] * emb[(size_t)(cb + j) * D_N + d];
    __syncthreads();
  }
  pooled[(size_t)s * HID_N + tid] = acc;  // tid == h*64 + d
}

// ---------------------------------------------------------------------------
// Kernel C: out[10000,128] = pooled[10000,256] @ W_out[256,128].
// One wave per 16x16 output tile, fp32 WMMA over K=256 (64 steps of 4).
// ---------------------------------------------------------------------------
__global__ __launch_bounds__(128) void mha_out_kernel(
    const float* __restrict__ pooled, const float* __restrict__ Wout,
    float* __restrict__ out) {
  const int tid  = threadIdx.x;
  const int wid  = blockIdx.x * 4 + (tid >> 5);
  const int lane = tid & 31;
  const int half = lane >> 4;
  const int l    = lane & 15;
  const int mt   = wid >> 3;  // 625 row tiles
  const int nt   = wid & 7;   // 8 col tiles
  const int col  = nt * 16 + l;

  const float* prow = pooled + (size_t)(mt * 16 + l) * HID_N;

  v8f c = {0.f, 0.f, 0.f, 0.f, 0.f, 0.f, 0.f, 0.f};
  for (int k = 0; k < 64; ++k) {
    const int kr = 4 * k + 2 * half;
    const v2f a = *reinterpret_cast<const v2f*>(prow + kr);
    v2f b;
    b.x = Wout[(size_t)kr * OUT_N + col];
    b.y = Wout[(size_t)(kr + 1) * OUT_N + col];
    c = __builtin_amdgcn_wmma_f32_16x16x4_f32(
        false, a, false, b, (short)0, c, false, false);
  }
#pragma unroll
  for (int r = 0; r < 8; ++r)
    out[(size_t)(mt * 16 + half * 8 + r) * OUT_N + nt * 16 + l] = c[r];
}

// ---------------------------------------------------------------------------
extern "C" void kernel_launch(void* const* d_in, const int* in_sizes, int n_in,
                              void* d_out, int out_size, void* d_ws, size_t ws_size,
                              hipStream_t stream) {
  const float* emb  = (const float*)d_in[0];  // [V, 64]
  const float* Wv   = (const float*)d_in[1];  // [64, 256]
  const float* Wout = (const float*)d_in[2];  // [256, 128]
  const int*   seg  = (const int*)d_in[3];    // [V], sorted
  // d_in[4]: num_samples scalar (compile-time constant S_N)

  float* scores = (float*)d_ws;                    // V*H   = 2.0M floats
  float* pooled = scores + (size_t)V_N * H_N;      // S*HID = 2.56M floats
  float* outp   = (float*)d_out;                   // [S, OUT]

  // Stage A: 7813 blocks x 4 waves x 16 rows (tail waves self-disable).
  mha_scores_kernel<<<(V_N + 63) / 64, 128, 0, stream>>>(emb, Wv, scores);

  // Stage B: one block per sample.
  mha_pool_kernel<<<S_N, 256, 0, stream>>>(emb, scores, seg, pooled);

  // Stage C: 625*8 = 5000 wave-tiles, 4 waves/block -> 1250 blocks.
  mha_out_kernel<<<(S_N / 16) * (OUT_N / 16) / 4, 128, 0, stream>>>(pooled, Wout, outp);
}